// GNNActorCritic_20332375179289
// MI455X (gfx1250) — compile-verified
//
#include <hip/hip_runtime.h>

typedef float v2f __attribute__((ext_vector_type(2)));
typedef float v8f __attribute__((ext_vector_type(8)));

#define CH 128  // IN_CH == HID_CH == 128

// ---------------- degree count: one thread per edge ----------------
__global__ void count_deg_kernel(const int* __restrict__ ei, float* __restrict__ cnt,
                                 int nEdges) {
    int e = blockIdx.x * blockDim.x + threadIdx.x;
    if (e >= nEdges) return;
    int dst = ei[nEdges + e];
    atomicAdd(cnt + dst, 1.0f);
}

// ------------- scatter-add: 32 threads/edge, float4 each -------------
__global__ void scatter_add_kernel(const float* __restrict__ feat,
                                   const int* __restrict__ ei,
                                   float* __restrict__ agg, int nEdges) {
    long long t = (long long)blockIdx.x * blockDim.x + threadIdx.x;
    int e = (int)(t >> 5);
    int part = (int)(t & 31);
    if (e >= nEdges) return;
    int src = ei[e];
    int dst = ei[nEdges + e];
    float4 v = ((const float4*)(feat + (size_t)src * CH))[part];
    float* base = agg + (size_t)dst * CH + part * 4;
    atomicAdd(base + 0, v.x);
    atomicAdd(base + 1, v.y);
    atomicAdd(base + 2, v.z);
    atomicAdd(base + 3, v.w);
}

// ------------- in-place mean: agg[n][:] /= max(cnt[n],1) -------------
__global__ void mean_kernel(float* __restrict__ agg, const float* __restrict__ cnt,
                            int nNodes) {
    int i = blockIdx.x * blockDim.x + threadIdx.x; // one float4 per thread
    if (i >= nNodes * (CH / 4)) return;
    int n = i >> 5;
    float c = fmaxf(cnt[n], 1.0f);
    float4* p = (float4*)agg + i;
    float4 v = *p;
    v.x /= c; v.y /= c; v.z /= c; v.w /= c;
    *p = v;
}

// ----------------------------------------------------------------------
// WMMA fused linear:  out = [relu]( A1 @ W1^T + bias (+ A2 @ W2^T) )
// A1,A2: [N, 128] row-major.  W1,W2: [OUTC, 128] row-major.  out: [N, OUTC].
// One wave per 16x16 output tile; blockDim.x = (OUTC/16)*32; grid.x = N/16.
// K = 128 fixed, stepped by 4 with V_WMMA_F32_16X16X4_F32.
// Safe for out == A1 (in-place): __syncthreads() separates reads from writes
// and each block only reads/writes its own 16 rows.
// ----------------------------------------------------------------------
__global__ void wmma_linear_kernel(const float* __restrict__ A1,
                                   const float* __restrict__ W1,
                                   const float* __restrict__ bias,
                                   const float* __restrict__ A2,
                                   const float* __restrict__ W2,
                                   float* __restrict__ out,
                                   int OUTC, int doRelu) {
    const int node0   = blockIdx.x * 16;
    const int lane    = threadIdx.x & 31;
    const int wave    = threadIdx.x >> 5;
    const int outbase = wave * 16;

    // A-matrix 16x4 layout: lanes 0-15 hold K={k,k+1}, lanes 16-31 hold K={k+2,k+3}
    const int row  = lane & 15;           // M for A, N for B/C/D (lanes mod 16)
    const int koff = (lane >> 4) << 1;    // 0 or 2

    const float* a1row = A1 + (size_t)(node0 + row) * CH + koff;
    const float* w1row = W1 + (size_t)(outbase + row) * CH + koff; // B col n = lane&15

    v8f acc = {};
    #pragma unroll
    for (int k = 0; k < CH; k += 4) {
        v2f a; a.x = a1row[k]; a.y = a1row[k + 1];
        v2f b; b.x = w1row[k]; b.y = w1row[k + 1];
        acc = __builtin_amdgcn_wmma_f32_16x16x4_f32(false, a, false, b,
                                                    (short)0, acc, false, false);
    }
    if (A2 != nullptr) {
        const float* a2row = A2 + (size_t)(node0 + row) * CH + koff;
        const float* w2row = W2 + (size_t)(outbase + row) * CH + koff;
        #pragma unroll
        for (int k = 0; k < CH; k += 4) {
            v2f a; a.x = a2row[k]; a.y = a2row[k + 1];
            v2f b; b.x = w2row[k]; b.y = w2row[k + 1];
            acc = __builtin_amdgcn_wmma_f32_16x16x4_f32(false, a, false, b,
                                                        (short)0, acc, false, false);
        }
    }

    const float bv = bias[outbase + row];   // per-column bias (col = lane&15)
    __syncthreads();                        // all reads done before in-place writes

    // C/D layout: VGPR r holds M = r + 8*(lane>=16), N = lane&15
    const int mbase = (lane >> 4) << 3;
    #pragma unroll
    for (int r = 0; r < 8; ++r) {
        float v = acc[r] + bv;
        if (doRelu) v = fmaxf(v, 0.0f);
        out[(size_t)(node0 + mbase + r) * OUTC + outbase + row] = v;
    }
}

// --------------- values head: per-node dot with wc + bc ---------------
__global__ void values_kernel(const float* __restrict__ h,
                              const float* __restrict__ wc,
                              const float* __restrict__ bc,
                              float* __restrict__ out, int nNodes) {
    int i = blockIdx.x * blockDim.x + threadIdx.x;
    if (i >= nNodes) return;
    const float4* hp = (const float4*)(h + (size_t)i * CH);
    const float4* wp = (const float4*)wc;
    float s = 0.0f;
    #pragma unroll
    for (int j = 0; j < CH / 4; ++j) {
        float4 a = hp[j], b = wp[j];
        s += a.x * b.x + a.y * b.y + a.z * b.z + a.w * b.w;
    }
    out[i] = s + bc[0];
}

extern "C" void kernel_launch(void* const* d_in, const int* in_sizes, int n_in,
                              void* d_out, int out_size, void* d_ws, size_t ws_size,
                              hipStream_t stream) {
    const float* x    = (const float*)d_in[0];
    const int*   ei   = (const int*)d_in[1];   // edge_index [2, E] (int32 on device)
    const float* w1l  = (const float*)d_in[2];
    const float* b1l  = (const float*)d_in[3];
    const float* w1r  = (const float*)d_in[4];
    const float* w2l  = (const float*)d_in[5];
    const float* b2l  = (const float*)d_in[6];
    const float* w2r  = (const float*)d_in[7];
    const float* wa   = (const float*)d_in[8];
    const float* ba   = (const float*)d_in[9];
    const float* wc   = (const float*)d_in[10];
    const float* bc   = (const float*)d_in[11];

    const int N = in_sizes[0] / CH;     // 40000
    const int E = in_sizes[1] / 2;      // 640000
    const int OUT = 64;

    // workspace layout (256B aligned): cnt[N] | agg[N*CH] | h1[N*CH]
    char*  ws   = (char*)d_ws;
    size_t cntB = (((size_t)N * 4) + 255) & ~(size_t)255;
    float* cnt  = (float*)ws;
    float* agg  = (float*)(ws + cntB);
    float* h1   = agg + (size_t)N * CH;

    float* logits = (float*)d_out;              // [N, 64]
    float* values = logits + (size_t)N * OUT;   // [N]

    const int T = 256;
    const int degBlocks  = (E + T - 1) / T;
    const int scatBlocks = (int)(((long long)E * 32 + T - 1) / T);
    const int meanBlocks = (N * (CH / 4) + T - 1) / T;
    const int tileBlocks = N / 16;              // 2500 (N multiple of 16)
    const int valBlocks  = (N + T - 1) / T;

    // ---- layer 1 aggregation ----
    hipMemsetAsync(cnt, 0, (size_t)N * 4, stream);
    hipMemsetAsync(agg, 0, (size_t)N * CH * 4, stream);
    count_deg_kernel<<<degBlocks, T, 0, stream>>>(ei, cnt, E);
    scatter_add_kernel<<<scatBlocks, T, 0, stream>>>(x, ei, agg, E);
    mean_kernel<<<meanBlocks, T, 0, stream>>>(agg, cnt, N);
    // h1 = relu(mean @ w1l^T + b1l + x @ w1r^T)
    wmma_linear_kernel<<<tileBlocks, (CH / 16) * 32, 0, stream>>>(
        agg, w1l, b1l, x, w1r, h1, CH, 1);

    // ---- layer 2 aggregation (degrees unchanged) ----
    hipMemsetAsync(agg, 0, (size_t)N * CH * 4, stream);
    scatter_add_kernel<<<scatBlocks, T, 0, stream>>>(h1, ei, agg, E);
    mean_kernel<<<meanBlocks, T, 0, stream>>>(agg, cnt, N);
    // h2 = relu(mean2 @ w2l^T + b2l + h1 @ w2r^T), written in place over agg
    wmma_linear_kernel<<<tileBlocks, (CH / 16) * 32, 0, stream>>>(
        agg, w2l, b2l, h1, w2r, agg, CH, 1);

    // ---- heads ----
    wmma_linear_kernel<<<tileBlocks, (OUT / 16) * 32, 0, stream>>>(
        agg, wa, ba, nullptr, nullptr, logits, OUT, 0);
    values_kernel<<<valBlocks, T, 0, stream>>>(agg, wc, bc, values, N);
}